// Laplace_Mean_Euclidean_Loss_23167053595271
// MI455X (gfx1250) — compile-verified
//
#include <hip/hip_runtime.h>

// Problem constants (match the reference)
#define POINT_NUM  50000
#define MAX_DEGREE 16
#define BATCH      32
#define NBLOCKS2   512   // partial count for stage-2 (8 chunks of 64 for WMMA finisher)

typedef __attribute__((ext_vector_type(2))) float v2f;
typedef __attribute__((ext_vector_type(8))) float v8f;

// ---------------------------------------------------------------------------
// Pass 1: Dt[(n*3 + c)*BATCH + b] = pred[b][n][c] - gt[b][n][c]; row n==POINT_NUM = 0.
// Output writes are perfectly coalesced (thread id == output index).
// Input reads are strided but the 38.4 MB of pred/gt live comfortably in the
// 192 MB L2, so effective HBM traffic stays ~= the data size.
// ---------------------------------------------------------------------------
__global__ void diff_transpose_kernel(const float* __restrict__ pred,
                                      const float* __restrict__ gt,
                                      float* __restrict__ Dt) {
  const int t = blockIdx.x * blockDim.x + threadIdx.x;
  const int total = (POINT_NUM + 1) * 3 * BATCH;
  if (t >= total) return;
  const int n   = t / (3 * BATCH);
  const int rem = t - n * (3 * BATCH);
  const int c   = rem >> 5;        // component 0..2
  const int b   = rem & 31;        // batch lane
  float v = 0.0f;
  if (n < POINT_NUM) {
    const size_t src = ((size_t)b * POINT_NUM + n) * 3 + c;
    v = pred[src] - gt[src];
  }
  Dt[t] = v;
}

// ---------------------------------------------------------------------------
// Pass 2: one wave32 per point (grid-stride); lane = batch index.
// The point index n is wave-uniform, so we force it scalar with
// readfirstlane: the 16 int64 neighbour indices and the degree then lower to
// SMEM scalar loads (broadcast for free, KMcnt-tracked), and all 48 gather
// addresses can be formed from SGPRs with no ds_bpermute chain in the loop.
// Each (neighbour, component) gather is one fully-coalesced 128B transaction
// from Dt, which at 19.2 MB stays resident in the 192 MB L2 — the kernel is
// L2-gather-bandwidth bound, which is the floor for this access pattern.
// ---------------------------------------------------------------------------
__global__ void laplace_loss_kernel(const float* __restrict__ Dt,
                                    const long long* __restrict__ nbr,
                                    const int* __restrict__ degrees,
                                    float* __restrict__ partials) {
  const int lane          = threadIdx.x & 31;
  const int waveInBlock   = threadIdx.x >> 5;
  const int wavesPerBlock = blockDim.x >> 5;
  const int waveId        = blockIdx.x * wavesPerBlock + waveInBlock;
  const int numWaves      = gridDim.x * wavesPerBlock;

  float acc = 0.0f;
  for (int n = waveId; n < POINT_NUM; n += numWaves) {
    // Wave-uniform point index -> scalar loads for indices + degree
    const int nu = __builtin_amdgcn_readfirstlane(n);
    const long long* __restrict__ row = nbr + (size_t)nu * MAX_DEGREE;
    const float deg = (float)degrees[nu];            // s_load + broadcast

    const size_t own = (size_t)nu * (3 * BATCH) + lane;
    float sx = deg * Dt[own];
    float sy = deg * Dt[own + BATCH];
    float sz = deg * Dt[own + 2 * BATCH];

#pragma unroll
    for (int d = 0; d < MAX_DEGREE; ++d) {
      const int idx = (int)row[d];                   // uniform -> SMEM load
      const size_t g = (size_t)idx * (3 * BATCH) + lane;
      sx -= Dt[g];                                   // coalesced 128B gathers
      sy -= Dt[g + BATCH];
      sz -= Dt[g + 2 * BATCH];
    }
    acc += sqrtf(sx * sx + sy * sy + sz * sz);
  }

  // Deterministic block reduction (fixed-order tree)
  __shared__ float red[256];
  red[threadIdx.x] = acc;
  __syncthreads();
  for (int s = blockDim.x >> 1; s > 0; s >>= 1) {
    if (threadIdx.x < s) red[threadIdx.x] += red[threadIdx.x + s];
    __syncthreads();
  }
  if (threadIdx.x == 0) partials[blockIdx.x] = red[0];
}

// ---------------------------------------------------------------------------
// Pass 3: single wave32 reduces the 512 f32 partials with chained
// V_WMMA_F32_16X16X4_F32. B = all-ones, so each WMMA accumulates 4-wide row
// sums into C; 8 chunks of 64 cover all 512 values in f32 precision with a
// hardware-fixed (deterministic) summation order. Final 16-row sum folds the
// C/D layout: VGPR r holds M=r (lanes 0-15) and M=8+r (lanes 16-31).
// EXEC is all-ones here (32 threads, no divergence before the WMMA).
// ---------------------------------------------------------------------------
__global__ void final_reduce_wmma(const float* __restrict__ partials,
                                  float* __restrict__ out) {
  const int lane  = threadIdx.x;        // 0..31
  const int M     = lane & 15;
  const int kbase = (lane >> 4) << 1;   // lanes 0-15: K={0,1}; lanes 16-31: K={2,3}

  v8f c = {};
  v2f ones;
  ones.x = 1.0f;
  ones.y = 1.0f;

#pragma unroll
  for (int i = 0; i < 8; ++i) {
    v2f a;
    a.x = partials[i * 64 + M * 4 + kbase];
    a.y = partials[i * 64 + M * 4 + kbase + 1];
    // D[m,n] = sum_k A[m,k] * 1 + C[m,n]  -> row sums accumulate in C
    c = __builtin_amdgcn_wmma_f32_16x16x4_f32(
        /*neg_a=*/false, a, /*neg_b=*/false, ones,
        /*c_mod=*/(short)0, c, /*reuse_a=*/false, /*reuse_b=*/false);
  }

  // Column N=lane of D is identical for every N (B was all-ones):
  // lanes 0-15 hold rows 0..7 in c[0..7], lanes 16-31 hold rows 8..15.
  float s = c[0] + c[1] + c[2] + c[3] + c[4] + c[5] + c[6] + c[7];
  const float tot = s + __shfl_xor(s, 16, 32);  // add the other half's rows
  if (lane == 0) out[0] = tot / (float)BATCH;   // mean over batch
}

// ---------------------------------------------------------------------------
extern "C" void kernel_launch(void* const* d_in, const int* in_sizes, int n_in,
                              void* d_out, int out_size, void* d_ws, size_t ws_size,
                              hipStream_t stream) {
  const float*      pred = (const float*)d_in[0];       // [B, N, 3] f32
  const float*      gt   = (const float*)d_in[1];       // [B, N, 3] f32
  const long long*  nbr  = (const long long*)d_in[2];   // [N, 16] i64
  const int*        deg  = (const int*)d_in[3];         // [N] i32

  // Workspace layout: Dt[(N+1)*3*B] floats, then NBLOCKS2 partial sums.
  float* Dt = (float*)d_ws;
  const size_t dtBytes = (size_t)(POINT_NUM + 1) * 3 * BATCH * sizeof(float); // 19,200,384 B
  float* partials = (float*)((char*)d_ws + dtBytes);

  const int totalDt = (POINT_NUM + 1) * 3 * BATCH;
  diff_transpose_kernel<<<(totalDt + 255) / 256, 256, 0, stream>>>(pred, gt, Dt);

  laplace_loss_kernel<<<NBLOCKS2, 256, 0, stream>>>(Dt, nbr, deg, partials);

  final_reduce_wmma<<<1, 32, 0, stream>>>(partials, (float*)d_out);
}